// SimpleGCNLayer_6382321402034
// MI455X (gfx1250) — compile-verified
//
#include <hip/hip_runtime.h>

#define D 128

typedef float v2f __attribute__((ext_vector_type(2)));
typedef float v8f __attribute__((ext_vector_type(8)));

// ---------------- Kernel 1: init deg=1 (self loop) and zero output accumulator
__global__ void __launch_bounds__(256) gcn_init_kernel(float* __restrict__ deg,
                                                       float* __restrict__ out,
                                                       long long N) {
    long long idx = (long long)blockIdx.x * blockDim.x + threadIdx.x;
    long long total = N * D;
    if (idx < total) out[idx] = 0.0f;
    if (idx < N) deg[idx] = 1.0f;   // self loop contributes 1 to every dst degree
}

// ---------------- Kernel 2: degree over edge destinations
__global__ void __launch_bounds__(256) gcn_degree_kernel(const long long* __restrict__ ei,
                                                         float* __restrict__ deg,
                                                         long long E) {
    long long e = (long long)blockIdx.x * blockDim.x + threadIdx.x;
    if (e < E) {
        long long d = ei[E + e];    // edge_index[1][e]
        atomicAdd(&deg[d], 1.0f);
    }
}

// ---------------- Kernel 3: dinv = rsqrt(deg) in place (deg >= 1 always)
__global__ void __launch_bounds__(256) gcn_rsqrt_kernel(float* __restrict__ deg, long long N) {
    long long i = (long long)blockIdx.x * blockDim.x + threadIdx.x;
    if (i < N) deg[i] = __frsqrt_rn(deg[i]);
}

// ---------------- Kernel 4: h = x @ W^T via V_WMMA_F32_16X16X4_F32
// Block = 256 threads = 8 waves. Block mt owns a 16-row slab of x (staged in LDS);
// wave w computes the 16x16 tile at columns [16w, 16w+16).
__global__ void __launch_bounds__(256) gcn_gemm_kernel(const float* __restrict__ x,
                                                       const float* __restrict__ W,
                                                       float* __restrict__ h,
                                                       long long N) {
    __shared__ float As[16 * D];            // 8 KB x-slab
    const long long mt   = blockIdx.x;
    const int       t    = threadIdx.x;
    const long long row0 = mt * 16;

    // cooperative load of 16x128 x tile (guard tail rows with zero)
    for (int i = t; i < 16 * D; i += 256) {
        long long gr = row0 + (i / D);
        As[i] = (gr < N) ? x[gr * D + (i % D)] : 0.0f;
    }
    __syncthreads();

    const int wave = t >> 5;                // output column tile (0..7)
    const int lane = t & 31;
    const int half = lane >> 4;             // 0: lanes 0-15, 1: lanes 16-31
    const int rc   = lane & 15;             // A row / B column within tile

    // B[k][n] = W^T[k][n] = W[n][k]  (W row-major [128][128])
    const float* wrow = W + (long long)(wave * 16 + rc) * D;

    v8f c = {0.f, 0.f, 0.f, 0.f, 0.f, 0.f, 0.f, 0.f};
    #pragma unroll
    for (int k0 = 0; k0 < D; k0 += 4) {
        const int k = k0 + 2 * half;        // lanes 0-15 -> K=k0,k0+1 ; 16-31 -> K=k0+2,k0+3
        v2f a; a.x = As[rc * D + k];  a.y = As[rc * D + k + 1];
        v2f b; b.x = wrow[k];         b.y = wrow[k + 1];
        c = __builtin_amdgcn_wmma_f32_16x16x4_f32(false, a, false, b,
                                                  (short)0, c, false, false);
    }

    // C/D layout: VGPR r -> M = r + 8*half, N = rc
    float* hb = h + row0 * D + wave * 16;
    #pragma unroll
    for (int r = 0; r < 8; ++r) {
        long long gr = row0 + r + 8 * half;
        if (gr < N) hb[(long long)(r + 8 * half) * D + rc] = c[r];
    }
}

// ---------------- Kernel 5: per-edge gather/scale/scatter-add (one wave32 per edge)
__global__ void __launch_bounds__(256) gcn_scatter_kernel(const long long* __restrict__ ei,
                                                          const float* __restrict__ dinv,
                                                          const float* __restrict__ h,
                                                          float* __restrict__ out,
                                                          long long E) {
    const int lane = threadIdx.x & 31;
    const long long e = (long long)blockIdx.x * 8 + (threadIdx.x >> 5);
    if (e >= E) return;
    const long long s = ei[e];
    const long long d = ei[E + e];
    const float norm = dinv[s] * dinv[d];

    float4 v = ((const float4*)(h + s * D))[lane];   // 128 floats = 32 lanes x float4
    float* ob = out + d * D + (long long)lane * 4;
    atomicAdd(ob + 0, v.x * norm);
    atomicAdd(ob + 1, v.y * norm);
    atomicAdd(ob + 2, v.z * norm);
    atomicAdd(ob + 3, v.w * norm);
}

// ---------------- Kernel 6: self-loop + bias + ReLU + LayerNorm, one wave32 per node
__global__ void __launch_bounds__(256) gcn_finalize_kernel(const float* __restrict__ h,
                                                           const float* __restrict__ dinv,
                                                           const float* __restrict__ bias,
                                                           const float* __restrict__ gamma,
                                                           const float* __restrict__ beta,
                                                           float* __restrict__ out,
                                                           long long N) {
    const int lane = threadIdx.x & 31;
    const long long n = (long long)blockIdx.x * 8 + (threadIdx.x >> 5);
    if (n >= N) return;

    const float di   = dinv[n];
    const float self = di * di;

    float4 acc = ((const float4*)(out + n * D))[lane];
    float4 hs  = ((const float4*)(h   + n * D))[lane];
    float4 bb  = ((const float4*)bias)[lane];

    float4 r;
    r.x = fmaxf(fmaf(hs.x, self, acc.x) + bb.x, 0.0f);
    r.y = fmaxf(fmaf(hs.y, self, acc.y) + bb.y, 0.0f);
    r.z = fmaxf(fmaf(hs.z, self, acc.z) + bb.z, 0.0f);
    r.w = fmaxf(fmaf(hs.w, self, acc.w) + bb.w, 0.0f);

    // mean over 128 features (wave32 butterfly)
    float s1 = r.x + r.y + r.z + r.w;
    #pragma unroll
    for (int off = 16; off > 0; off >>= 1) s1 += __shfl_xor(s1, off, 32);
    const float mu = s1 * (1.0f / 128.0f);

    float dx0 = r.x - mu, dx1 = r.y - mu, dx2 = r.z - mu, dx3 = r.w - mu;
    float s2 = dx0 * dx0 + dx1 * dx1 + dx2 * dx2 + dx3 * dx3;
    #pragma unroll
    for (int off = 16; off > 0; off >>= 1) s2 += __shfl_xor(s2, off, 32);
    const float inv = __frsqrt_rn(s2 * (1.0f / 128.0f) + 1e-5f);

    float4 g  = ((const float4*)gamma)[lane];
    float4 be = ((const float4*)beta)[lane];
    float4 o;
    o.x = dx0 * inv * g.x + be.x;
    o.y = dx1 * inv * g.y + be.y;
    o.z = dx2 * inv * g.z + be.z;
    o.w = dx3 * inv * g.w + be.w;
    ((float4*)(out + n * D))[lane] = o;
}

extern "C" void kernel_launch(void* const* d_in, const int* in_sizes, int n_in,
                              void* d_out, int out_size, void* d_ws, size_t ws_size,
                              hipStream_t stream) {
    const float*     x     = (const float*)d_in[0];
    const long long* ei    = (const long long*)d_in[1];   // int64 [2,E]
    const float*     W     = (const float*)d_in[2];
    const float*     bias  = (const float*)d_in[3];
    const float*     gamma = (const float*)d_in[4];
    const float*     beta  = (const float*)d_in[5];
    float*           out   = (float*)d_out;

    const long long N = (long long)in_sizes[0] / D;
    const long long E = (long long)in_sizes[1] / 2;

    // workspace layout: [0,N) dinv/deg ; [N, N + N*D) h
    float* deg = (float*)d_ws;
    float* h   = deg + N;

    {   // init: zero out accumulator, deg = 1 (self loop)
        long long total = N * D;
        int blocks = (int)((total + 255) / 256);
        gcn_init_kernel<<<blocks, 256, 0, stream>>>(deg, out, N);
    }
    {   // degree
        int blocks = (int)((E + 255) / 256);
        gcn_degree_kernel<<<blocks, 256, 0, stream>>>(ei, deg, E);
    }
    {   // dinv
        int blocks = (int)((N + 255) / 256);
        gcn_rsqrt_kernel<<<blocks, 256, 0, stream>>>(deg, N);
    }
    {   // WMMA GEMM: one block per 16-row slab, 8 waves cover all 8 column tiles
        int blocks = (int)((N + 15) / 16);
        gcn_gemm_kernel<<<blocks, 256, 0, stream>>>(x, W, h, N);
    }
    {   // edge scatter: one wave per edge, 8 waves per block
        int blocks = (int)((E + 7) / 8);
        gcn_scatter_kernel<<<blocks, 256, 0, stream>>>(ei, deg, h, out, E);
    }
    {   // finalize: one wave per node
        int blocks = (int)((N + 7) / 8);
        gcn_finalize_kernel<<<blocks, 256, 0, stream>>>(h, deg, bias, gamma, beta, out, N);
    }
}